// TwoStageRegressionLoss_80341658239297
// MI455X (gfx1250) — compile-verified
//
#include <hip/hip_runtime.h>
#include <hip/hip_bf16.h>

typedef float v2f __attribute__((ext_vector_type(2)));
typedef float v4f __attribute__((ext_vector_type(4)));
typedef float v8f __attribute__((ext_vector_type(8)));

#define NBLOCKS 2048
#define NTHREADS 256
#define NWAVES (NTHREADS / 32)

__device__ __forceinline__ void accum_elem(float bpv, float btv, float rpv, float rtv,
                                           float& bsum, float& rsum) {
    // ---- break loss: torch-style BCE with log clamp at -100 ----
    float p = __saturatef(bpv);           // clip(pred, 0, 1)
    float t = __saturatef(btv);           // clip(target, 0, 1)
    // Hardware v_log_f32 (fast path). log1p(-p) ~= log(1-p): for p in [0,1]
    // the 1-p rounding error is <=2^-24 absolute -> negligible in a mean loss.
    float lp  = fmaxf(__logf(p),        -100.0f);
    float l1p = fmaxf(__logf(1.0f - p), -100.0f);
    // -(t*lp + (1-t)*l1p) = -(l1p + t*(lp - l1p))
    bsum -= fmaf(t, lp - l1p, l1p);

    // ---- regression: duration-aware focal smooth-L1 ----
    float d = fabsf(rpv - rtv);
    float u = fminf(d * 2.0f, 1.0f);      // clip(d/DELTA,0,1), DELTA=0.5
    float focal = u * u;                  // GAMMA=2
    float base = (d < 0.5f) ? (d * d) : (d - 0.25f);   // 0.5*d*d/DELTA : d-0.5*DELTA
    float c = rintf(rtv);                 // round-half-even, matches jnp.round
    c = fminf(fmaxf(c, 0.0f), 3.0f);      // -> v_med3
    float w = (c == 0.0f) ? 1.0f : (c == 1.0f) ? 4.0f : (c == 2.0f) ? 3.0f : 2.0f;
    rsum = fmaf(0.25f * focal * base, w, rsum);        // ALPHA=0.25
}

__global__ void __launch_bounds__(NTHREADS)
loss_main(const v4f* __restrict__ bp4, const v4f* __restrict__ rp4,
          const v4f* __restrict__ bt4, const v4f* __restrict__ rt4,
          float* __restrict__ partB, float* __restrict__ partR,
          int nChunks) {
    const int tid    = blockIdx.x * blockDim.x + threadIdx.x;
    const int stride = gridDim.x * blockDim.x;

    float bsum = 0.0f, rsum = 0.0f;

    // Streaming phase: 128-bit non-temporal loads (537 MB > 192 MB L2,
    // touched once -> NT hint avoids cache pollution). Unroll 4 => up to 16
    // outstanding b128 loads per thread for latency hiding (LOADcnt max 63).
    #pragma unroll 4
    for (int i = tid; i < nChunks; i += stride) {
        v4f bp = __builtin_nontemporal_load(&bp4[i]);
        v4f rp = __builtin_nontemporal_load(&rp4[i]);
        v4f bt = __builtin_nontemporal_load(&bt4[i]);
        v4f rt = __builtin_nontemporal_load(&rt4[i]);
        accum_elem(bp.x, bt.x, rp.x, rt.x, bsum, rsum);
        accum_elem(bp.y, bt.y, rp.y, rt.y, bsum, rsum);
        accum_elem(bp.z, bt.z, rp.z, rt.z, bsum, rsum);
        accum_elem(bp.w, bt.w, rp.w, rt.w, bsum, rsum);
    }

    // ---- Wave reduction on the matrix pipe: V_WMMA_F32_16X16X4_F32 ----
    // A (16x4 f32): lane L<16 holds A[L][0]=VGPR0, A[L][1]=VGPR1;
    //               lane L>=16 holds A[L-16][2]=VGPR0, A[L-16][3]=VGPR1.
    // Put break partial in slots K={0,2}, reg partial in slots K={1,3}.
    // B (4x16 f32): column 0 = ones over K={0,2}; column 1 = ones over K={1,3}.
    //   VGPR0: lanes 0-15 = B[0][n], lanes 16-31 = B[2][n]
    //   VGPR1: lanes 0-15 = B[1][n], lanes 16-31 = B[3][n]
    const unsigned lane = threadIdx.x & 31u;
    v2f a;  a.x = bsum; a.y = rsum;
    v2f bb;
    bb.x = (lane == 0u || lane == 16u) ? 1.0f : 0.0f;  // break -> column n=0
    bb.y = (lane == 1u || lane == 17u) ? 1.0f : 0.0f;  // reg   -> column n=1
    v8f cz = {};
    v8f d = __builtin_amdgcn_wmma_f32_16x16x4_f32(
        /*neg_a=*/false, a, /*neg_b=*/false, bb,
        /*c_mod=*/(short)0, cz, /*reuse_a=*/false, /*reuse_b=*/false);
    // D[m][n] lives in lane (n + 16*(m>=8)), VGPR (m%8).
    float s = d[0] + d[1] + d[2] + d[3] + d[4] + d[5] + d[6] + d[7];
    // lanes 0,16: two halves of wave break sum; lanes 1,17: reg sum halves.

    __shared__ float sB[2 * NWAVES];
    __shared__ float sR[2 * NWAVES];
    const int wv = threadIdx.x >> 5;
    if      (lane == 0u)  sB[2 * wv + 0] = s;
    else if (lane == 16u) sB[2 * wv + 1] = s;
    else if (lane == 1u)  sR[2 * wv + 0] = s;
    else if (lane == 17u) sR[2 * wv + 1] = s;
    __syncthreads();

    if (threadIdx.x == 0) {
        float b = 0.0f, r = 0.0f;
        #pragma unroll
        for (int i = 0; i < 2 * NWAVES; ++i) { b += sB[i]; r += sR[i]; }
        partB[blockIdx.x] = b;
        partR[blockIdx.x] = r;
    }
}

__global__ void __launch_bounds__(NTHREADS)
loss_final(const float* __restrict__ partB, const float* __restrict__ partR,
           float* __restrict__ out, int nPart, float invN) {
    float b = 0.0f, r = 0.0f;
    for (int i = threadIdx.x; i < nPart; i += blockDim.x) {
        b += partB[i];
        r += partR[i];
    }
    #pragma unroll
    for (int off = 16; off > 0; off >>= 1) {
        b += __shfl_down(b, off, 32);
        r += __shfl_down(r, off, 32);
    }
    __shared__ float sb[NWAVES];
    __shared__ float sr[NWAVES];
    const int wv = threadIdx.x >> 5;
    if ((threadIdx.x & 31) == 0) { sb[wv] = b; sr[wv] = r; }
    __syncthreads();
    if (threadIdx.x == 0) {
        float B = 0.0f, R = 0.0f;
        #pragma unroll
        for (int i = 0; i < NWAVES; ++i) { B += sb[i]; R += sr[i]; }
        B *= invN;
        R *= invN;
        out[0] = B + R;   // total
        out[1] = B;       // break_loss
        out[2] = R;       // regression_loss
    }
}

extern "C" void kernel_launch(void* const* d_in, const int* in_sizes, int n_in,
                              void* d_out, int out_size, void* d_ws, size_t ws_size,
                              hipStream_t stream) {
    const float* bp = (const float*)d_in[0];  // break_predictions
    const float* rp = (const float*)d_in[1];  // regression_predictions
    const float* bt = (const float*)d_in[2];  // break_targets
    const float* rt = (const float*)d_in[3];  // regression_targets

    const int n = in_sizes[0];        // 2048*16384 = 33,554,432 (divisible by 4)
    const int nChunks = n >> 2;

    float* partB = (float*)d_ws;
    float* partR = partB + NBLOCKS;

    loss_main<<<NBLOCKS, NTHREADS, 0, stream>>>(
        (const v4f*)bp, (const v4f*)rp, (const v4f*)bt, (const v4f*)rt,
        partB, partR, nChunks);

    loss_final<<<1, NTHREADS, 0, stream>>>(
        partB, partR, (float*)d_out, NBLOCKS, 1.0f / (float)n);
}